// HypergraphConv_85521388798293
// MI455X (gfx1250) — compile-verified
//
#include <hip/hip_runtime.h>

// ---------------------------------------------------------------------------
// HypergraphConv for MI455X (gfx1250, wave32).
// Problem constants from the reference:
//   B=8, S=8192, Fin=Fout=128, E=65536 hyperedge entries, NUM_EDGES=8192.
// Phase 1: xt = x @ W  via V_WMMA_F32_16X16X4_F32 (full f32 precision; the
//          problem is memory/atomic bound, so fp32 matrix path is free).
// Phase 2: per-edge counts + per-edge sums of xt via hardware f32 atomics.
// Phase 3: best[node] = segment-max of eligible (count>1) edge ids.
// Phase 4: gather means / passthrough, write agg, reduce mean|agg - xt|.
// ---------------------------------------------------------------------------

#define B_DIM   8
#define S_DIM   8192
#define F_DIM   128
#define NEDGES  8192
#define N_OUT   ((size_t)B_DIM * S_DIM * F_DIM)   // 8388608

typedef float v2f __attribute__((ext_vector_type(2)));
typedef float v8f __attribute__((ext_vector_type(8)));

// ---------------------------------------------------------------------------
// Kernel 0: zero the accumulator state (sums, counts, best, loss slot).
// Grid sized exactly to NEDGES*B_DIM*F_DIM elements (= 8388608).
// ---------------------------------------------------------------------------
__global__ __launch_bounds__(256) void init_ws_kernel(float* __restrict__ sums,
                                                      int* __restrict__ counts,
                                                      int* __restrict__ best,
                                                      float* __restrict__ lossSlot) {
  const size_t i = (size_t)blockIdx.x * blockDim.x + threadIdx.x;
  sums[i] = 0.0f;
  if (i < NEDGES) {
    counts[i] = 0;
    best[i]   = -1;
  }
  if (i == 0) *lossSlot = 0.0f;
}

// ---------------------------------------------------------------------------
// Kernel 1: xt[M=65536, 128] = x[M,128] @ W[128,128] with f32 WMMA.
// 256 threads = 8 waves per block; block owns a 16-row slab, each wave owns
// one 16-wide N tile. K loop = 32 x V_WMMA_F32_16X16X4_F32 per wave.
//
// VGPR layouts per CDNA5 ISA 7.12.2 (wave32):
//   A (16x4 f32):  lane L -> row M=L%16, VGPR v holds K = 2*(L/16) + v
//   B (4x16 f32):  lane L -> col N=L%16, VGPR v holds K = 2*(L/16) + v
//   C/D (16x16):   lane L, VGPR v -> row M = v + 8*(L/16), col N = L%16
// EXEC is all-1s: grid is exact, no divergence around the WMMA.
// ---------------------------------------------------------------------------
__global__ __launch_bounds__(256) void gemm_xt_wmma(const float* __restrict__ x,
                                                    const float* __restrict__ w,
                                                    float* __restrict__ xt) {
  const int lane    = threadIdx.x & 31;
  const int wave    = threadIdx.x >> 5;      // 0..7 -> N tile
  const int rowBase = blockIdx.x << 4;       // 16 rows per block
  const int mIdx    = lane & 15;
  const int kHalf   = lane >> 4;             // 0 or 1
  const int nBase   = wave << 4;

  const float* xRow = x + (size_t)(rowBase + mIdx) * F_DIM;
  const float* wCol = w + nBase + mIdx;

  v8f acc = {};
#pragma unroll
  for (int k0 = 0; k0 < F_DIM; k0 += 4) {
    const int k = k0 + (kHalf << 1);
    const v2f a = *(const v2f*)(xRow + k);          // global_load_b64
    v2f b;
    b.x = wCol[(size_t)k * F_DIM];                  // W row-major: stride 128
    b.y = wCol[(size_t)(k + 1) * F_DIM];
    acc = __builtin_amdgcn_wmma_f32_16x16x4_f32(
        /*neg_a=*/false, a, /*neg_b=*/false, b,
        /*c_mod=*/(short)0, acc, /*reuse_a=*/false, /*reuse_b=*/false);
  }

  float* dst = xt + (size_t)(rowBase + (kHalf << 3)) * F_DIM + nBase + mIdx;
#pragma unroll
  for (int v = 0; v < 8; ++v)
    dst[(size_t)v * F_DIM] = acc[v];
}

// ---------------------------------------------------------------------------
// Kernel 2: one block per hyperedge entry. counts[edge]++ and
// sums[edge,b,f] += xt[b,node,f] for all (b,f) — 1024 hardware f32 atomics
// per entry, fully coalesced in both the gather and the scatter.
// ---------------------------------------------------------------------------
__global__ __launch_bounds__(256) void edge_accum_kernel(const int* __restrict__ node_idx,
                                                         const int* __restrict__ edge_idx,
                                                         const float* __restrict__ xt,
                                                         float* __restrict__ sums,
                                                         int* __restrict__ counts) {
  const int e    = blockIdx.x;
  const int node = node_idx[e];
  const int edge = edge_idx[e];
  if (threadIdx.x == 0) atomicAdd(&counts[edge], 1);

#pragma unroll
  for (int i = threadIdx.x; i < B_DIM * F_DIM; i += 256) {
    const int b = i >> 7;
    const int f = i & (F_DIM - 1);
    const float val = xt[((size_t)b * S_DIM + node) * F_DIM + f];
    // Agent-scope relaxed fadd -> global_atomic_add_f32 (no CAS loop).
    __hip_atomic_fetch_add(&sums[((size_t)edge * B_DIM + b) * F_DIM + f], val,
                           __ATOMIC_RELAXED, __HIP_MEMORY_SCOPE_AGENT);
  }
}

// ---------------------------------------------------------------------------
// Kernel 3: best[node] = max edge id among entries whose edge has count > 1.
// ---------------------------------------------------------------------------
__global__ __launch_bounds__(256) void best_edge_kernel(const int* __restrict__ node_idx,
                                                        const int* __restrict__ edge_idx,
                                                        const int* __restrict__ counts,
                                                        int* __restrict__ best,
                                                        int E) {
  const int e = blockIdx.x * blockDim.x + threadIdx.x;
  if (e >= E) return;
  const int edge = edge_idx[e];
  if (counts[edge] > 1) atomicMax(&best[node_idx[e]], edge);
}

// ---------------------------------------------------------------------------
// Kernel 4: agg = (best>=0) ? sums[best]/count[best] : xt; write agg and
// accumulate mean|agg - xt| (pre-scaled by 1/N) via LDS tree + one atomic.
// idx bit layout: [b(3) | n(13) | f(7)].
// ---------------------------------------------------------------------------
__global__ __launch_bounds__(256) void finalize_kernel(const float* __restrict__ xt,
                                                       const float* __restrict__ sums,
                                                       const int* __restrict__ counts,
                                                       const int* __restrict__ best,
                                                       float* __restrict__ out,
                                                       float* __restrict__ lossSlot) {
  __shared__ float red[256];
  const size_t idx = (size_t)blockIdx.x * blockDim.x + threadIdx.x;
  const int f = (int)(idx & (F_DIM - 1));
  const int n = (int)((idx >> 7) & (S_DIM - 1));
  const int b = (int)(idx >> 20);

  const float base = xt[idx];
  const int be = best[n];
  float v = base;
  if (be >= 0) {
    const int c = counts[be];
    const float denom = (float)(c > 1 ? c : 1);
    v = sums[((size_t)be * B_DIM + b) * F_DIM + f] / denom;
  }
  out[idx] = v;

  const float invN = 1.0f / (float)N_OUT;
  red[threadIdx.x] = fabsf(v - base) * invN;
  __syncthreads();
#pragma unroll
  for (int s = 128; s > 0; s >>= 1) {
    if ((int)threadIdx.x < s) red[threadIdx.x] += red[threadIdx.x + s];
    __syncthreads();
  }
  if (threadIdx.x == 0)
    __hip_atomic_fetch_add(lossSlot, red[0], __ATOMIC_RELAXED,
                           __HIP_MEMORY_SCOPE_AGENT);
}

// ---------------------------------------------------------------------------
extern "C" void kernel_launch(void* const* d_in, const int* in_sizes, int n_in,
                              void* d_out, int out_size, void* d_ws, size_t ws_size,
                              hipStream_t stream) {
  const float* x    = (const float*)d_in[0];
  const int*   hidx = (const int*)d_in[1];     // [2, E] int32 (JAX x64-off)
  const float* w    = (const float*)d_in[2];

  const int E = in_sizes[1] / 2;
  const int* node_idx = hidx;
  const int* edge_idx = hidx + E;

  // Workspace layout (bytes): xt 32MB | sums 32MB | counts 32KB | best 32KB
  char*  ws     = (char*)d_ws;
  float* xt     = (float*)ws;
  float* sums   = (float*)(ws + N_OUT * sizeof(float));
  int*   counts = (int*)(ws + 2 * N_OUT * sizeof(float));
  int*   best   = (int*)(ws + 2 * N_OUT * sizeof(float) + NEDGES * sizeof(int));

  float* out      = (float*)d_out;
  float* lossSlot = out + N_OUT;     // out_size == N_OUT + 1

  const int sumsElems = NEDGES * B_DIM * F_DIM;       // == N_OUT

  init_ws_kernel<<<sumsElems / 256, 256, 0, stream>>>(sums, counts, best, lossSlot);

  gemm_xt_wmma<<<(B_DIM * S_DIM) / 16, 256, 0, stream>>>(x, w, xt);

  edge_accum_kernel<<<E, 256, 0, stream>>>(node_idx, edge_idx, xt, sums, counts);

  best_edge_kernel<<<(E + 255) / 256, 256, 0, stream>>>(node_idx, edge_idx,
                                                        counts, best, E);

  finalize_kernel<<<(int)(N_OUT / 256), 256, 0, stream>>>(xt, sums, counts, best,
                                                          out, lossSlot);
}